// TabGraph_48086453846268
// MI455X (gfx1250) — compile-verified
//
#include <hip/hip_runtime.h>
#include <hip/hip_bf16.h>
#include <math.h>

typedef __attribute__((ext_vector_type(2))) float v2f;
typedef __attribute__((ext_vector_type(8))) float v8f;

#define BB   8192        // graphs / batch
#define NPG  32          // nodes per graph
#define EPG  64          // edges per graph
#define NT   (BB*NPG)    // 262144 nodes
#define EE   (BB*EPG)    // 524288 edges
#define CELL 908
#define BN_EPS 1e-5f

// ---------------------------------------------------------------------------
// Generic fp32 GEMM via V_WMMA_F32_16X16X4_F32.
// Y[M,N] = X[M,K] @ W[K,N] + bias[N]
// Each wave computes a 16x32 tile (two 16x16 WMMA tiles sharing the A frag);
// block = 8 waves stacked along M -> block tile 128x32.
// Fast path (wave-uniform "tile fully in bounds"): unguarded loads, zero
// masking overhead. Edge path: clamped addresses + value masks (v_cndmask),
// never conditional loads, so EXEC stays all-ones around the WMMAs.
// A 16x4 layout: lanes 0-15 -> rows, {v0,v1}={K0,K1}; lanes 16-31 -> {K2,K3}.
// B 4x16 layout: lanes 0-15 -> cols, {v0,v1}={K0,K1}; lanes 16-31 -> {K2,K3}.
// ---------------------------------------------------------------------------
__global__ __launch_bounds__(256)
void wmma_gemm_f32(const float* __restrict__ X, const float* __restrict__ W,
                   const float* __restrict__ bias, float* __restrict__ Y,
                   int M, int N, int K)
{
    const int wave = threadIdx.x >> 5;
    const int lane = threadIdx.x & 31;
    const int n0 = blockIdx.x * 32;
    const int m0 = (blockIdx.y * 8 + wave) * 16;
    if (m0 >= M) return;                 // uniform per wave -> EXEC stays full

    const int half = lane >> 4;          // 0: K+0/1, 1: K+2/3
    const int l16  = lane & 15;
    const int am   = m0 + l16;           // A row (M is a multiple of 16 here)
    const int bn0  = n0 + l16;           // B col, tile 0
    const int bn1  = n0 + 16 + l16;      // B col, tile 1
    const size_t arow = (size_t)am * K;

    v8f acc0 = {}, acc1 = {};

    if (n0 + 32 <= N) {
        // -------- fast path: whole 32-wide tile in bounds (wave-uniform) ----
        const float* __restrict__ W0 = W + bn0;
        const float* __restrict__ W1 = W + bn1;
        int k0 = 0;
#pragma unroll 4
        for (; k0 + 4 <= K; k0 += 4) {
            const int ka = k0 + half * 2;
            v2f a, b0, b1;
            a.x = X[arow + ka];
            a.y = X[arow + ka + 1];
            b0.x = W0[(size_t)ka * N];
            b0.y = W0[(size_t)(ka + 1) * N];
            b1.x = W1[(size_t)ka * N];
            b1.y = W1[(size_t)(ka + 1) * N];
            acc0 = __builtin_amdgcn_wmma_f32_16x16x4_f32(false, a, false, b0,
                                                         (short)0, acc0, false, false);
            acc1 = __builtin_amdgcn_wmma_f32_16x16x4_f32(false, a, false, b1,
                                                         (short)0, acc1, false, false);
        }
        if (k0 < K) {                    // K tail (K % 4 != 0)
            const int ka  = k0 + half * 2;
            const int ka0 = (ka     < K) ? ka     : (K - 1);
            const int ka1 = (ka + 1 < K) ? ka + 1 : (K - 1);
            const bool kv0 = (ka < K), kv1 = (ka + 1 < K);
            v2f a, b0, b1;
            const float ax  = X[arow + ka0];
            const float ay  = X[arow + ka1];
            const float t00 = W0[(size_t)ka0 * N];
            const float t01 = W0[(size_t)ka1 * N];
            const float t10 = W1[(size_t)ka0 * N];
            const float t11 = W1[(size_t)ka1 * N];
            a.x  = kv0 ? ax  : 0.f;  a.y  = kv1 ? ay  : 0.f;
            b0.x = kv0 ? t00 : 0.f;  b0.y = kv1 ? t01 : 0.f;
            b1.x = kv0 ? t10 : 0.f;  b1.y = kv1 ? t11 : 0.f;
            acc0 = __builtin_amdgcn_wmma_f32_16x16x4_f32(false, a, false, b0,
                                                         (short)0, acc0, false, false);
            acc1 = __builtin_amdgcn_wmma_f32_16x16x4_f32(false, a, false, b1,
                                                         (short)0, acc1, false, false);
        }
        const float bv0 = bias[bn0];
        const float bv1 = bias[bn1];
        const int mrow = m0 + half * 8;
#pragma unroll
        for (int r = 0; r < 8; ++r) {
            const int m = mrow + r;
            if (m < M) {
                Y[(size_t)m * N + bn0] = acc0[r] + bv0;
                Y[(size_t)m * N + bn1] = acc1[r] + bv1;
            }
        }
    } else {
        // -------- edge path: clamp addresses, mask values ------------------
        const bool v0  = (bn0 < N);
        const bool v1  = (bn1 < N);
        const int  bc0 = v0 ? bn0 : (N - 1);
        const int  bc1 = v1 ? bn1 : (N - 1);
        const float* __restrict__ W0 = W + bc0;
        const float* __restrict__ W1 = W + bc1;
        int k0 = 0;
#pragma unroll 4
        for (; k0 + 4 <= K; k0 += 4) {
            const int ka = k0 + half * 2;
            v2f a, b0, b1;
            a.x = X[arow + ka];
            a.y = X[arow + ka + 1];
            const float t00 = W0[(size_t)ka * N];
            const float t01 = W0[(size_t)(ka + 1) * N];
            const float t10 = W1[(size_t)ka * N];
            const float t11 = W1[(size_t)(ka + 1) * N];
            b0.x = v0 ? t00 : 0.f;  b0.y = v0 ? t01 : 0.f;
            b1.x = v1 ? t10 : 0.f;  b1.y = v1 ? t11 : 0.f;
            acc0 = __builtin_amdgcn_wmma_f32_16x16x4_f32(false, a, false, b0,
                                                         (short)0, acc0, false, false);
            acc1 = __builtin_amdgcn_wmma_f32_16x16x4_f32(false, a, false, b1,
                                                         (short)0, acc1, false, false);
        }
        if (k0 < K) {
            const int ka  = k0 + half * 2;
            const int ka0 = (ka     < K) ? ka     : (K - 1);
            const int ka1 = (ka + 1 < K) ? ka + 1 : (K - 1);
            const bool kv0 = (ka < K), kv1 = (ka + 1 < K);
            v2f a, b0, b1;
            const float ax  = X[arow + ka0];
            const float ay  = X[arow + ka1];
            const float t00 = W0[(size_t)ka0 * N];
            const float t01 = W0[(size_t)ka1 * N];
            const float t10 = W1[(size_t)ka0 * N];
            const float t11 = W1[(size_t)ka1 * N];
            a.x  = kv0 ? ax : 0.f;          a.y  = kv1 ? ay : 0.f;
            b0.x = (v0 && kv0) ? t00 : 0.f; b0.y = (v0 && kv1) ? t01 : 0.f;
            b1.x = (v1 && kv0) ? t10 : 0.f; b1.y = (v1 && kv1) ? t11 : 0.f;
            acc0 = __builtin_amdgcn_wmma_f32_16x16x4_f32(false, a, false, b0,
                                                         (short)0, acc0, false, false);
            acc1 = __builtin_amdgcn_wmma_f32_16x16x4_f32(false, a, false, b1,
                                                         (short)0, acc1, false, false);
        }
        const float bt0 = bias[bc0];
        const float bt1 = bias[bc1];
        const float bv0 = v0 ? bt0 : 0.f;
        const float bv1 = v1 ? bt1 : 0.f;
        const int mrow = m0 + half * 8;
#pragma unroll
        for (int r = 0; r < 8; ++r) {
            const int m = mrow + r;
            if (m < M) {
                if (v0) Y[(size_t)m * N + bn0] = acc0[r] + bv0;
                if (v1) Y[(size_t)m * N + bn1] = acc1[r] + bv1;
            }
        }
    }
}

// ---------------------------------------------------------------------------
// Column statistics: stats[c] += sum, stats[N+c] += sumsq  (stats pre-zeroed)
// ---------------------------------------------------------------------------
__global__ __launch_bounds__(128)
void colstats(const float* __restrict__ X, float* __restrict__ stats,
              int M, int N, int rowsPerBlock)
{
    const int c = blockIdx.x * blockDim.x + threadIdx.x;
    if (c >= N) return;
    const int r0 = blockIdx.y * rowsPerBlock;
    const int r1 = min(M, r0 + rowsPerBlock);
    float s = 0.f, q = 0.f;
    for (int r = r0; r < r1; ++r) {
        const float v = X[(size_t)r * N + c];
        s += v; q += v * v;
    }
    atomicAdd(&stats[c], s);
    atomicAdd(&stats[N + c], q);
}

__global__ __launch_bounds__(128)
void bn_finalize(float* __restrict__ stats, int N, float invM)
{
    const int c = blockIdx.x * blockDim.x + threadIdx.x;
    if (c >= N) return;
    const float mean = stats[c] * invM;
    const float var  = stats[N + c] * invM - mean * mean;
    stats[c]     = mean;
    stats[N + c] = rsqrtf(var + BN_EPS);
}

// ---------------------------------------------------------------------------
// Fused (optional BN) + activation + strided/offset store (for concat writes)
// act: 0=none, 1=relu, 2=elu
// ---------------------------------------------------------------------------
__global__ __launch_bounds__(256)
void bn_act(const float* __restrict__ in, const float* __restrict__ stats,
            float* __restrict__ out, int M, int N, int outStride, int colOff,
            int act, int useBN)
{
    const size_t i = (size_t)blockIdx.x * blockDim.x + threadIdx.x;
    if (i >= (size_t)M * N) return;
    const int r = (int)(i / N);
    const int c = (int)(i % N);
    float v = in[i];
    if (useBN) v = (v - stats[c]) * stats[N + c];
    if (act == 1)       v = fmaxf(v, 0.f);
    else if (act == 2)  v = (v > 0.f) ? v : expm1f(v);
    out[(size_t)r * outStride + colOff + c] = v;
}

// ---------------------------------------------------------------------------
// Zero-pad 9x128 weight to 12x128 so the GIN-1 GEMM has K % 4 == 0
// ---------------------------------------------------------------------------
__global__ __launch_bounds__(256)
void pad_w9(const float* __restrict__ w, float* __restrict__ wp)
{
    const int i = blockIdx.x * blockDim.x + threadIdx.x;
    if (i >= 12 * 128) return;
    const int k = i / 128, n = i % 128;
    wp[i] = (k < 9) ? w[k * 128 + n] : 0.f;
}

// ---------------------------------------------------------------------------
// GIN aggregation, feat=9 input -> zero-padded N x 12 output: h0 = x + agg(x)
// One block per graph; LDS accumulation with ds_add_f32 atomics.
// ---------------------------------------------------------------------------
__global__ __launch_bounds__(128)
void gin_agg9(const float* __restrict__ x, const int* __restrict__ src,
              const int* __restrict__ dst, float* __restrict__ h0)
{
    __shared__ float xs[NPG * 12];
    __shared__ float acc[NPG * 12];
    const int g = blockIdx.x, tid = threadIdx.x;
    for (int i = tid; i < NPG * 12; i += 128) { xs[i] = 0.f; acc[i] = 0.f; }
    __syncthreads();
    for (int i = tid; i < NPG * 9; i += 128) {
        const int n = i / 9, f = i % 9;
        xs[n * 12 + f] = x[((size_t)g * NPG + n) * 9 + f];
    }
    __syncthreads();
    for (int i = tid; i < EPG * 9; i += 128) {
        const int e = i / 9, f = i % 9;
        const int sl = src[g * EPG + e] & (NPG - 1);
        const int dl = dst[g * EPG + e] & (NPG - 1);
        atomicAdd(&acc[dl * 12 + f], xs[sl * 12 + f]);
    }
    __syncthreads();
    for (int i = tid; i < NPG * 12; i += 128)
        h0[(size_t)g * NPG * 12 + i] = xs[i] + acc[i];
}

// ---------------------------------------------------------------------------
// GIN aggregation, feat=128: out = h + agg(h). One block per graph, 32 KB LDS.
// ---------------------------------------------------------------------------
__global__ __launch_bounds__(256)
void gin_agg128(const float* __restrict__ h, const int* __restrict__ src,
                const int* __restrict__ dst, float* __restrict__ out)
{
    __shared__ float hs[NPG * 128];
    __shared__ float acc[NPG * 128];
    const int g = blockIdx.x, tid = threadIdx.x;
    for (int i = tid; i < NPG * 128; i += 256) {
        hs[i]  = h[(size_t)g * NPG * 128 + i];
        acc[i] = 0.f;
    }
    __syncthreads();
    for (int i = tid; i < EPG * 128; i += 256) {
        const int e = i >> 7, f = i & 127;
        const int sl = src[g * EPG + e] & (NPG - 1);
        const int dl = dst[g * EPG + e] & (NPG - 1);
        atomicAdd(&acc[dl * 128 + f], hs[sl * 128 + f]);
    }
    __syncthreads();
    for (int i = tid; i < NPG * 128; i += 256)
        out[(size_t)g * NPG * 128 + i] = hs[i] + acc[i];
}

// ---------------------------------------------------------------------------
// Per-graph max pool over 32 nodes, 128 features. Block per graph.
// ---------------------------------------------------------------------------
__global__ __launch_bounds__(128)
void pool_max(const float* __restrict__ h, float* __restrict__ p)
{
    const int g = blockIdx.x, f = threadIdx.x;
    float m = -INFINITY;
#pragma unroll
    for (int n = 0; n < NPG; ++n)
        m = fmaxf(m, h[(size_t)g * NPG * 128 + n * 128 + f]);
    p[(size_t)g * 128 + f] = m;
}

// ---------------------------------------------------------------------------
// Final 64 -> 1 projection
// ---------------------------------------------------------------------------
__global__ __launch_bounds__(256)
void final_dot(const float* __restrict__ f2, const float* __restrict__ w,
               const float* __restrict__ b, float* __restrict__ y, int M)
{
    const int r = blockIdx.x * blockDim.x + threadIdx.x;
    if (r >= M) return;
    float s = b[0];
#pragma unroll
    for (int k = 0; k < 64; ++k) s += f2[(size_t)r * 64 + k] * w[k];
    y[r] = s;
}

// ---------------------------------------------------------------------------
// Host-side sequencing
// ---------------------------------------------------------------------------
static inline void gemm(hipStream_t s, const float* X, const float* W,
                        const float* b, float* Y, int M, int N, int K)
{
    dim3 grid((N + 31) / 32, (M + 127) / 128);
    wmma_gemm_f32<<<grid, 256, 0, s>>>(X, W, b, Y, M, N, K);
}

static inline void bn_stats(hipStream_t s, float* stats, const float* X,
                            int M, int N)
{
    hipMemsetAsync(stats, 0, 2 * (size_t)N * sizeof(float), s);
    dim3 g((N + 127) / 128, (M + 2047) / 2048);
    colstats<<<g, 128, 0, s>>>(X, stats, M, N, 2048);
    bn_finalize<<<(N + 127) / 128, 128, 0, s>>>(stats, N, 1.0f / (float)M);
}

static inline void act_apply(hipStream_t s, const float* in, const float* stats,
                             float* out, int M, int N, int outStride, int colOff,
                             int act, int useBN)
{
    const size_t total = (size_t)M * N;
    bn_act<<<(unsigned)((total + 255) / 256), 256, 0, s>>>(
        in, stats, out, M, N, outStride, colOff, act, useBN);
}

extern "C" void kernel_launch(void* const* d_in, const int* in_sizes, int n_in,
                              void* d_out, int out_size, void* d_ws, size_t ws_size,
                              hipStream_t stream)
{
    const float* cell   = (const float*)d_in[0];
    const float* drug_x = (const float*)d_in[1];
    const int*   eidx   = (const int*)d_in[2];
    const int*   src    = eidx;
    const int*   dst    = eidx + EE;
    // d_in[3] = drug_batch (implicit: node/NPG)
    const float *ce1w = (const float*)d_in[4],  *ce1b = (const float*)d_in[5];
    const float *ce2w = (const float*)d_in[6],  *ce2b = (const float*)d_in[7];
    const float *ce3w = (const float*)d_in[8],  *ce3b = (const float*)d_in[9];
    const float *g11w = (const float*)d_in[10], *g11b = (const float*)d_in[11];
    const float *g12w = (const float*)d_in[12], *g12b = (const float*)d_in[13];
    const float *g21w = (const float*)d_in[14], *g21b = (const float*)d_in[15];
    const float *g22w = (const float*)d_in[16], *g22b = (const float*)d_in[17];
    const float *d1w  = (const float*)d_in[18], *d1b  = (const float*)d_in[19];
    const float *d2w  = (const float*)d_in[20], *d2b  = (const float*)d_in[21];
    const float *f1w  = (const float*)d_in[22], *f1b  = (const float*)d_in[23];
    const float *f2w  = (const float*)d_in[24], *f2b  = (const float*)d_in[25];
    const float *f3w  = (const float*)d_in[26], *f3b  = (const float*)d_in[27];
    float* y = (float*)d_out;

    // Workspace layout (floats); R1/R2 are NT x 128 and are aggressively aliased.
    float* ws = (float*)d_ws;
    const size_t SZ_BIG = (size_t)NT * 128;        // 33.55M floats
    float* R1    = ws;                             // big region 1
    float* R2    = R1 + SZ_BIG;                    // big region 2
    float* Z     = R2 + SZ_BIG;                    // [B,256] concat buffer
    float* P     = Z + (size_t)BB * 256;           // [B,128] pooled
    float* STATS = P + (size_t)BB * 128;           // 2*516 max
    float* G1WP  = STATS + 2048;                   // 12x128 padded weight

    // ---- cell branch (uses R1/R2 as scratch before the GIN phase) ----
    float* c1 = R1;                                // [B,516]
    float* c2 = R2;                                // [B,256]
    float* c3 = R1;                                // [B,128] (c1 dead)
    gemm(stream, cell, ce1w, ce1b, c1, BB, 516, CELL);
    bn_stats(stream, STATS, c1, BB, 516);
    act_apply(stream, c1, STATS, c1, BB, 516, 516, 0, /*relu*/1, /*bn*/1);
    gemm(stream, c1, ce2w, ce2b, c2, BB, 256, 516);
    bn_stats(stream, STATS, c2, BB, 256);
    act_apply(stream, c2, STATS, c2, BB, 256, 256, 0, 1, 1);
    gemm(stream, c2, ce3w, ce3b, c3, BB, 128, 256);
    bn_stats(stream, STATS, c3, BB, 128);
    act_apply(stream, c3, STATS, Z, BB, 128, 256, 0, 1, 1);   // Z[:, :128] = c

    // ---- GIN conv 1 ----
    pad_w9<<<(12 * 128 + 255) / 256, 256, 0, stream>>>(g11w, G1WP);
    float* H0 = R2;                                // [NT,12] (c2 dead)
    gin_agg9<<<BB, 128, 0, stream>>>(drug_x, src, dst, H0);
    gemm(stream, H0, G1WP, g11b, R1, NT, 128, 12); // t1 = h0 @ W1 + b
    bn_stats(stream, STATS, R1, NT, 128);
    act_apply(stream, R1, STATS, R1, NT, 128, 128, 0, 1, 1);  // relu(bn(.))
    gemm(stream, R1, g12w, g12b, R2, NT, 128, 128);           // t2 (H0 dead)
    act_apply(stream, R2, STATS, R2, NT, 128, 128, 0, 1, 0);  // relu only
    bn_stats(stream, STATS, R2, NT, 128);
    act_apply(stream, R2, STATS, R2, NT, 128, 128, 0, 0, 1);  // bn -> h1 in R2

    // ---- GIN conv 2 ----
    gin_agg128<<<BB, 256, 0, stream>>>(R2, src, dst, R1);     // hp = h1+agg -> R1
    gemm(stream, R1, g21w, g21b, R2, NT, 128, 128);           // (h1 dead)
    bn_stats(stream, STATS, R2, NT, 128);
    act_apply(stream, R2, STATS, R2, NT, 128, 128, 0, 1, 1);
    gemm(stream, R2, g22w, g22b, R1, NT, 128, 128);
    act_apply(stream, R1, STATS, R1, NT, 128, 128, 0, 1, 0);  // relu
    bn_stats(stream, STATS, R1, NT, 128);
    act_apply(stream, R1, STATS, R1, NT, 128, 128, 0, 0, 1);  // bn -> h2 in R1

    // ---- pool + drug MLP ----
    pool_max<<<BB, 128, 0, stream>>>(R1, P);
    float* D1 = R2;                                // [B,128]
    float* D2 = R2 + (size_t)BB * 128;
    gemm(stream, P, d1w, d1b, D1, BB, 128, 128);
    bn_stats(stream, STATS, D1, BB, 128);
    act_apply(stream, D1, STATS, D1, BB, 128, 128, 0, 1, 1);
    gemm(stream, D1, d2w, d2b, D2, BB, 128, 128);
    act_apply(stream, D2, STATS, Z, BB, 128, 256, 128, 1, 0); // Z[:,128:] = relu(d2)

    // ---- fusion head ----
    float* F1 = R2 + 2 * (size_t)BB * 128;
    float* F2 = R2 + 3 * (size_t)BB * 128;
    gemm(stream, Z, f1w, f1b, F1, BB, 128, 256);
    bn_stats(stream, STATS, F1, BB, 128);
    act_apply(stream, F1, STATS, F1, BB, 128, 128, 0, /*elu*/2, 1);
    gemm(stream, F1, f2w, f2b, F2, BB, 64, 128);
    bn_stats(stream, STATS, F2, BB, 64);
    act_apply(stream, F2, STATS, F2, BB, 64, 64, 0, 2, 1);
    final_dot<<<(BB + 255) / 256, 256, 0, stream>>>(F2, f3w, f3b, y, BB);
}